// RankingLoss_26508538151261
// MI455X (gfx1250) — compile-verified
//
#include <hip/hip_runtime.h>

typedef __attribute__((ext_vector_type(2))) float v2f;
typedef __attribute__((ext_vector_type(8))) float v8f;

#define RB 64
#define RN 2048
#define NTILES (RN / 16)          // 128 j-tiles per row
#define TOL    0.01f
#define LOG2E  1.44269504088896340736f
#define LN2    0.69314718055994530942f

__global__ void RankingLoss_init_kernel(float* out) {
    out[0] = 0.0f;
}

// One block = one (batch row, group of 8 i-tiles). 8 waves per block, each
// wave owns a 16-row i-tile and sweeps all 128 16-col j-tiles of that row.
__global__ __launch_bounds__(256) void RankingLoss_26508538151261_kernel(
    const float* __restrict__ logits,
    const float* __restrict__ labels,
    float* __restrict__ out)
{
    __shared__ float s_logit[RN];
    __shared__ float s_label[RN];
    __shared__ float s_red[8];

    const int b    = blockIdx.x >> 4;   // batch row (64)
    const int iblk = blockIdx.x & 15;   // which group of 8 i-tiles (16)
    const int tid  = threadIdx.x;

    const float* rowLogit = logits + (size_t)b * RN;
    const float* rowLabel = labels + (size_t)b * RN;

    // Stage the whole row into LDS once (16 KB of 320 KB/WGP).
    for (int k = tid; k < RN; k += 256) {
        s_logit[k] = rowLogit[k];
        s_label[k] = rowLabel[k];
    }
    __syncthreads();

    const int  wave = tid >> 5;
    const int  lane = tid & 31;
    const bool lo   = lane < 16;        // lanes 0-15 carry K=0,1; 16-31 carry K=2,3 (zeroed)
    const int  i0   = (iblk * 8 + wave) * 16;

    // A (16x4, f32): column K=0 = x_i[m], K=1 = 1, K=2..3 = 0.
    // Layout (ISA 7.12.2): lanes 0-15 hold M=0..15 with VGPR0=K0, VGPR1=K1;
    // lanes 16-31 hold K2/K3 -> zero so they don't contribute.
    const float li  = lo ? s_logit[i0 + (lane & 15)] : 0.0f;
    const float lbi = lo ? s_label[i0 + (lane & 15)] : 0.0f;
    const float one = lo ? 1.0f : 0.0f;
    v2f a_logit = { li,  one };
    v2f a_label = { lbi, one };

    float acc = 0.0f;

    for (int jt = 0; jt < NTILES; ++jt) {
        const int j  = jt * 16 + (lane & 15);
        const float lj  = s_logit[j];
        const float lbj = s_label[j];
        // B (4x16, f32): row K=0 = 1, K=1 = -y_j[n], K=2..3 = 0.
        v2f b_logit = { one, lo ? -lj  : 0.0f };
        v2f b_label = { one, lo ? -lbj : 0.0f };

        v8f zero = {};
        // D[m,n] = x_i[m] - y_j[n]  (exact f32 rank-2 outer combination)
        v8f dlogit = __builtin_amdgcn_wmma_f32_16x16x4_f32(
            false, a_logit, false, b_logit, (short)0, zero, false, false);
        v8f dlabel = __builtin_amdgcn_wmma_f32_16x16x4_f32(
            false, a_label, false, b_label, (short)0, zero, false, false);

        // C/D layout: VGPR r, lanes 0-15 -> (M=r, N=lane); lanes 16-31 -> (M=r+8).
        // Every lane's 8 values are real pairs of this 16x16 tile.
#pragma unroll
        for (int r = 0; r < 8; ++r) {
            const float x = dlogit[r];
            // log_sigmoid(x) = min(x,0) - ln2 * log2(1 + 2^(-|x|*log2e))
            const float t  = __builtin_amdgcn_exp2f(-__builtin_fabsf(x * LOG2E));
            const float lg = __builtin_amdgcn_logf(1.0f + t);
            const float c  = __builtin_fminf(x, 0.0f) - LN2 * lg;
            acc += (dlabel[r] > TOL) ? c : 0.0f;
        }
    }

    // Wave reduce (wave32) then block reduce, one atomic per block.
#pragma unroll
    for (int off = 16; off > 0; off >>= 1)
        acc += __shfl_xor(acc, off, 32);
    if (lane == 0) s_red[wave] = acc;
    __syncthreads();
    if (tid == 0) {
        float s = 0.0f;
#pragma unroll
        for (int w = 0; w < 8; ++w) s += s_red[w];
        atomicAdd(out, s * (1.0f / (float)RB));   // mean over batch
    }
}

extern "C" void kernel_launch(void* const* d_in, const int* in_sizes, int n_in,
                              void* d_out, int out_size, void* d_ws, size_t ws_size,
                              hipStream_t stream) {
    const float* logits = (const float*)d_in[0];
    const float* labels = (const float*)d_in[1];
    float* out = (float*)d_out;

    RankingLoss_init_kernel<<<1, 1, 0, stream>>>(out);
    // 64 rows * 16 i-tile-groups = 1024 blocks of 256 threads (8 waves each)
    RankingLoss_26508538151261_kernel<<<RB * 16, 256, 0, stream>>>(logits, labels, out);
}